// SAGE_EP_64312840290338
// MI455X (gfx1250) — compile-verified
//
#include <hip/hip_runtime.h>
#include <hip/hip_bf16.h>

// ---------------------------------------------------------------------------
// GraphSAGE forward for MI455X (gfx1250, wave32).
// - Dense GEMMs: native fp32 tensor path V_WMMA_F32_16X16X4_F32 with two
//   interleaved independent accumulator chains (Wl and Wr paths); weight
//   panels staged to LDS via async global->LDS (ASYNCcnt).
// - Edge aggregation: float4 gathers + relaxed agent-scope fp32 atomics
//   (z/agg buffers are L2-resident: <=102 MB each vs 192 MB L2).
// ---------------------------------------------------------------------------

typedef float v2f __attribute__((ext_vector_type(2)));
typedef float v8f __attribute__((ext_vector_type(8)));
typedef int   sage_b128 __attribute__((vector_size(16)));   // matches builtin param

#define SAGE_N 100000
#define SAGE_E 800000
#define SAGE_L 200000

#if defined(__has_builtin)
#if __has_builtin(__builtin_amdgcn_global_load_async_to_lds_b128)
#define SAGE_HAVE_ASYNC_LDS 1
#endif
#endif

// stage 16 contiguous bytes global -> LDS
__device__ __forceinline__ void sage_stage16(const float* __restrict__ gp,
                                             float* __restrict__ lp)
{
#ifdef SAGE_HAVE_ASYNC_LDS
    __builtin_amdgcn_global_load_async_to_lds_b128(
        (sage_b128*)gp, (sage_b128*)lp, 0, 0);
#else
    *(float4*)lp = *(const float4*)gp;
#endif
}

__device__ __forceinline__ void sage_stage_wait()
{
#ifdef SAGE_HAVE_ASYNC_LDS
#if __has_builtin(__builtin_amdgcn_s_wait_asynccnt)
    __builtin_amdgcn_s_wait_asynccnt(0);
#else
    asm volatile("s_wait_asynccnt 0x0" ::: "memory");
#endif
#endif
    __syncthreads();
}

// ---------------------------------------------------------------------------
// z0[n, 0:128] = emb[x_idx[n], 0:128]   (float4 per thread)
// ---------------------------------------------------------------------------
__global__ void __launch_bounds__(256)
sage_gather(const int* __restrict__ x_idx, const float* __restrict__ emb,
            float* __restrict__ z, int n)
{
    int idx = blockIdx.x * blockDim.x + threadIdx.x;   // n * 32 chunks of 4
    if (idx >= n * 32) return;
    int node = idx >> 5;
    int c    = idx & 31;
    int v = x_idx[node];
    *(float4*)(z + (size_t)node * 128 + c * 4) =
        *(const float4*)(emb + (size_t)v * 128 + c * 4);
}

// ---------------------------------------------------------------------------
// deg accumulation (int atomics) and reciprocal
// ---------------------------------------------------------------------------
__global__ void __launch_bounds__(256)
sage_deg_accum(const int* __restrict__ dst, int* __restrict__ deg, int e)
{
    int idx = blockIdx.x * blockDim.x + threadIdx.x;
    if (idx >= e) return;
    atomicAdd(deg + dst[idx], 1);
}

__global__ void __launch_bounds__(256)
sage_deg_inv(const int* __restrict__ deg, float* __restrict__ dinv, int n)
{
    int idx = blockIdx.x * blockDim.x + threadIdx.x;
    if (idx >= n) return;
    int d = deg[idx];
    dinv[idx] = 1.0f / (float)(d > 0 ? d : 1);
}

// ---------------------------------------------------------------------------
// agg[dst] += z[src] over all edges; one thread per (edge, 4-feature chunk).
// ---------------------------------------------------------------------------
__global__ void __launch_bounds__(256)
sage_scatter(const int* __restrict__ src, const int* __restrict__ dst,
             const float* __restrict__ zin, float* __restrict__ agg,
             int din, long long total)
{
    long long idx = (long long)blockIdx.x * blockDim.x + threadIdx.x;
    if (idx >= total) return;
    int chunks = din >> 2;
    int e = (int)(idx / chunks);
    int c = (int)(idx % chunks);
    int s = src[e];
    int d = dst[e];
    const float4 v = *(const float4*)(zin + (size_t)s * din + 4 * c);
    float* p = agg + (size_t)d * din + 4 * c;
    __hip_atomic_fetch_add(p + 0, v.x, __ATOMIC_RELAXED, __HIP_MEMORY_SCOPE_AGENT);
    __hip_atomic_fetch_add(p + 1, v.y, __ATOMIC_RELAXED, __HIP_MEMORY_SCOPE_AGENT);
    __hip_atomic_fetch_add(p + 2, v.z, __ATOMIC_RELAXED, __HIP_MEMORY_SCOPE_AGENT);
    __hip_atomic_fetch_add(p + 3, v.w, __ATOMIC_RELAXED, __HIP_MEMORY_SCOPE_AGENT);
}

// ---------------------------------------------------------------------------
// zout = [relu]( (agg @ Wl) * dinv + bl + zin @ Wr )
// Block = 256 threads = 8 waves; each wave owns a 16x16 output tile of the
// same 16-column panel (weight staging amortized 8x). Weight panels
// (din x 16 of Wl and Wr) are staged to LDS once per block via async
// global->LDS; each merged K-step issues two WMMAs into independent
// accumulator chains (accL for Wl path, accR for Wr path) so the tensor
// pipe always has a ready instruction.
// Fragment layouts per ISA 7.12.2:
//   A 16x4: lanes 0-15 M=0..15 K={0,1}; lanes 16-31 same M, K={2,3}
//   B 4x16: v0 = B[K=2*hi][N=lane&15], v1 = B[K=2*hi+1][N]
//   C/D: acc[r] -> row = r + 8*(lane>>4), col = lane&15
// ---------------------------------------------------------------------------
__global__ void __launch_bounds__(256)
sage_gemm_wmma(const float* __restrict__ agg, const float* __restrict__ dinv,
               const float* __restrict__ zin, const float* __restrict__ Wl,
               const float* __restrict__ bl, const float* __restrict__ Wr,
               float* __restrict__ zout, int din, int dout, int do_relu, int n)
{
    __shared__ float lWl[256 * 16];   // 16 KB
    __shared__ float lWr[256 * 16];   // 16 KB

    const int tid     = threadIdx.x;
    const int lane    = tid & 31;
    const int wave    = tid >> 5;
    const int laneLo  = lane & 15;
    const int laneHi  = lane >> 4;          // 0 or 1
    const int rowBase = (blockIdx.x * 8 + wave) * 16;
    const int colBase = blockIdx.y * 16;

    // ---- stage both weight panels: din rows x 16 cols, float4 chunks ----
    {
        const int nchunk = din * 4;          // din*16/4 float4 chunks per panel
        for (int i = tid; i < nchunk; i += 256) {
            const int k = i >> 2;
            const int c = (i & 3) * 4;
            sage_stage16(Wl + (size_t)k * dout + colBase + c, lWl + k * 16 + c);
            sage_stage16(Wr + (size_t)k * dout + colBase + c, lWr + k * 16 + c);
        }
    }
    sage_stage_wait();

    // tail block: whole waves skip uniformly (EXEC stays all-ones for WMMA)
    if (rowBase >= n) return;

    v8f accL;                                // agg @ Wl   (scaled by dinv later)
    v8f accR;                                // zin @ Wr + bias
    const float bias = bl[colBase + laneLo];
#pragma unroll
    for (int r = 0; r < 8; ++r) { accL[r] = 0.0f; accR[r] = bias; }

    const float* arow = agg + (size_t)(rowBase + laneLo) * din;
    const float* zrow = zin + (size_t)(rowBase + laneLo) * din;
    const int    col  = colBase + laneLo;

    for (int k = 0; k < din; k += 4) {
        const int ka = k + 2 * laneHi;
        v2f aL = *(const v2f*)(arow + ka);   // 8B-aligned (din % 4 == 0)
        v2f aR = *(const v2f*)(zrow + ka);
        v2f bL, bR;
        bL.x = lWl[(ka    ) * 16 + laneLo];
        bL.y = lWl[(ka + 1) * 16 + laneLo];
        bR.x = lWr[(ka    ) * 16 + laneLo];
        bR.y = lWr[(ka + 1) * 16 + laneLo];
        accL = __builtin_amdgcn_wmma_f32_16x16x4_f32(
                   false, aL, false, bL, (short)0, accL, false, false);
        accR = __builtin_amdgcn_wmma_f32_16x16x4_f32(
                   false, aR, false, bR, (short)0, accR, false, false);
    }

#pragma unroll
    for (int r = 0; r < 8; ++r) {
        const int row = rowBase + r + 8 * laneHi;
        float v = accL[r] * dinv[row] + accR[r];
        if (do_relu) v = fmaxf(v, 0.0f);
        zout[(size_t)row * dout + col] = v;
    }
}

// ---------------------------------------------------------------------------
// out[l] = dot(z[a_l], z[b_l]) over 128 features; one wave32 per pair.
// ---------------------------------------------------------------------------
__global__ void __launch_bounds__(256)
sage_edge_score(const int* __restrict__ ai, const int* __restrict__ bi,
                const float* __restrict__ z, float* __restrict__ out, int l)
{
    int gid  = blockIdx.x * blockDim.x + threadIdx.x;
    int wave = gid >> 5;
    int lane = gid & 31;
    if (wave >= l) return;
    int a = ai[wave];
    int b = bi[wave];
    const float4 va = *(const float4*)(z + (size_t)a * 128 + lane * 4);
    const float4 vb = *(const float4*)(z + (size_t)b * 128 + lane * 4);
    float s = va.x * vb.x + va.y * vb.y + va.z * vb.z + va.w * vb.w;
#pragma unroll
    for (int m = 16; m > 0; m >>= 1) s += __shfl_xor(s, m, 32);
    if (lane == 0) out[wave] = s;
}

// ---------------------------------------------------------------------------
extern "C" void kernel_launch(void* const* d_in, const int* in_sizes, int n_in,
                              void* d_out, int out_size, void* d_ws, size_t ws_size,
                              hipStream_t stream)
{
    (void)in_sizes; (void)n_in; (void)out_size; (void)ws_size;

    const int N = SAGE_N, E = SAGE_E, L = SAGE_L;
    const int DIMS[5] = {128, 256, 192, 128, 128};

    const int*   x_idx = (const int*)d_in[0];
    const int*   src   = (const int*)d_in[1];          // edge_index row 0
    const int*   dst   = ((const int*)d_in[1]) + E;    // edge_index row 1
    const int*   eli0  = (const int*)d_in[2];          // edge_label_index row 0
    const int*   eli1  = ((const int*)d_in[2]) + L;    // edge_label_index row 1
    const float* emb   = (const float*)d_in[3];
    const float* Wl[4] = {(const float*)d_in[4],  (const float*)d_in[7],
                          (const float*)d_in[10], (const float*)d_in[13]};
    const float* bl[4] = {(const float*)d_in[5],  (const float*)d_in[8],
                          (const float*)d_in[11], (const float*)d_in[14]};
    const float* Wr[4] = {(const float*)d_in[6],  (const float*)d_in[9],
                          (const float*)d_in[12], (const float*)d_in[15]};
    float* out = (float*)d_out;

    // workspace layout
    float* bufA = (float*)d_ws;                        // N x 256
    float* bufB = bufA + (size_t)N * 256;              // N x 256
    float* agg  = bufB + (size_t)N * 256;              // N x 256
    float* dinv = agg  + (size_t)N * 256;              // N floats
    int*   degi = (int*)(dinv + N);                    // N ints

    // 1. gather embeddings into bufA
    sage_gather<<<(N * 32 + 255) / 256, 256, 0, stream>>>(x_idx, emb, bufA, N);

    // 2. degrees -> 1/max(deg,1)
    (void)hipMemsetAsync(degi, 0, (size_t)N * sizeof(int), stream);
    sage_deg_accum<<<(E + 255) / 256, 256, 0, stream>>>(dst, degi, E);
    sage_deg_inv<<<(N + 255) / 256, 256, 0, stream>>>(degi, dinv, N);

    // 3. four SAGE layers (ping-pong bufA/bufB)
    float* zin  = bufA;
    float* zout = bufB;
    const int rowTiles = N / 16;                       // 6250, exact
    for (int i = 0; i < 4; ++i) {
        const int din  = DIMS[i];
        const int dout = DIMS[i + 1];

        (void)hipMemsetAsync(agg, 0, (size_t)N * din * sizeof(float), stream);

        long long total = (long long)E * (din >> 2);
        sage_scatter<<<(unsigned)((total + 255) / 256), 256, 0, stream>>>(
            src, dst, zin, agg, din, total);

        dim3 grid((rowTiles + 7) / 8, dout / 16);
        sage_gemm_wmma<<<grid, 256, 0, stream>>>(
            agg, dinv, zin, Wl[i], bl[i], Wr[i], zout, din, dout,
            (i < 3) ? 1 : 0, N);

        float* t = zin; zin = zout; zout = t;
    }
    // after 4 swaps final activations are in zin (== bufA)

    // 4. edge scores: one wave per label pair
    long long score_threads = (long long)L * 32;
    sage_edge_score<<<(unsigned)((score_threads + 255) / 256), 256, 0, stream>>>(
        eli0, eli1, zin, out, L);
}